// simclr_17386027614471
// MI455X (gfx1250) — compile-verified
//
#include <hip/hip_runtime.h>
#include <hip/hip_bf16.h>

typedef __attribute__((ext_vector_type(2))) float v2f;
typedef __attribute__((ext_vector_type(8))) float v8f;

#define N_NODES 100000
#define N_EDGES 1600000
#define FEAT    128
#define DIM     64
#define EMB     192
#define GRAPHS  1000
#define BN_EPS  1e-5f

// ---------------------------------------------------------------- utilities
__global__ void zero_kernel(float* __restrict__ p, int n) {
    int i = blockIdx.x * blockDim.x + threadIdx.x;
    if (i < n) p[i] = 0.0f;
}

__global__ void copy4_kernel(const float4* __restrict__ in, float4* __restrict__ out, int n4) {
    int i = blockIdx.x * blockDim.x + threadIdx.x;
    if (i < n4) out[i] = in[i];
}

// ------------------------------------------------------------- edge scatter
// agg[dst] += h[src] ; agg pre-initialized to h (self term, GIN eps=0).
// Consecutive threads cover consecutive features of one edge -> coalesced
// gather and coalesced f32 atomics (L2-resident tables on MI455X's 192MB L2).
template<int F>
__global__ __launch_bounds__(256) void scatter_add_kernel(
    const float* __restrict__ h, const int* __restrict__ src,
    const int* __restrict__ dst, float* __restrict__ agg, long long total)
{
    long long idx = (long long)blockIdx.x * 256 + threadIdx.x;
    if (idx >= total) return;
    int f = (int)(idx % F);
    long long e = idx / F;
    int s = src[e], d = dst[e];
    atomicAdd(&agg[(size_t)d * F + f], h[(size_t)s * F + f]);
}

// ------------------------------------------------- fused GIN MLP (WMMA f32)
// z = ReLU( ReLU( agg @ w1 + b1 ) @ w2 + b2 )  for 16 nodes per block,
// plus per-feature sum / sum-of-squares atomics for BatchNorm statistics.
// 4 waves per block; wave w owns output columns [16w, 16w+16).
template<int FIN>
__global__ __launch_bounds__(128) void gin_mlp_kernel(
    const float* __restrict__ agg, const float* __restrict__ w1,
    const float* __restrict__ b1,  const float* __restrict__ w2,
    const float* __restrict__ b2,  float* __restrict__ z,
    float* __restrict__ stats)
{
    const int XP = FIN + 4;          // padded LDS strides (avoid 64-bank conflicts)
    const int ZP = DIM + 4;
    __shared__ float lx[16 * (FIN + 4)];
    __shared__ float lz[16 * (DIM + 4)];

    const int tid  = threadIdx.x;
    const int lane = tid & 31;
    const int wave = tid >> 5;        // 0..3
    const int row0 = blockIdx.x * 16; // N_NODES % 16 == 0

    for (int i = tid; i < 16 * FIN; i += 128) {
        int r = i / FIN, c = i % FIN;
        lx[r * XP + c] = agg[(size_t)(row0 + r) * FIN + c];
    }
    __syncthreads();

    const int m  = lane & 15;   // A row / B,C column within tile
    const int hi = lane >> 4;   // lane half selects K (A/B) or M+8 (C)
    const int n0 = wave * 16;
    const int k0 = hi ? 2 : 0, k1 = hi ? 3 : 1;

    // ---- GEMM 1: (16 x FIN) @ (FIN x 64), K-tiled by 4 (V_WMMA_F32_16X16X4_F32)
    v8f c1 = {};
    for (int k = 0; k < FIN; k += 4) {
        v2f a, b;
        a.x = lx[m * XP + k + k0];
        a.y = lx[m * XP + k + k1];
        b.x = w1[(k + k0) * DIM + n0 + m];
        b.y = w1[(k + k1) * DIM + n0 + m];
        c1 = __builtin_amdgcn_wmma_f32_16x16x4_f32(false, a, false, b,
                                                   (short)0, c1, false, false);
    }
    float bias1 = b1[n0 + m];
#pragma unroll
    for (int v = 0; v < 8; ++v)
        lz[(v + hi * 8) * ZP + n0 + m] = fmaxf(c1[v] + bias1, 0.0f);
    __syncthreads();

    // ---- GEMM 2: (16 x 64) @ (64 x 64)
    v8f c2 = {};
    for (int k = 0; k < DIM; k += 4) {
        v2f a, b;
        a.x = lz[m * ZP + k + k0];
        a.y = lz[m * ZP + k + k1];
        b.x = w2[(k + k0) * DIM + n0 + m];
        b.y = w2[(k + k1) * DIM + n0 + m];
        c2 = __builtin_amdgcn_wmma_f32_16x16x4_f32(false, a, false, b,
                                                   (short)0, c2, false, false);
    }
    float bias2 = b2[n0 + m];
    float csum = 0.0f, csq = 0.0f;
#pragma unroll
    for (int v = 0; v < 8; ++v) {
        float val = fmaxf(c2[v] + bias2, 0.0f);
        z[(size_t)(row0 + v + hi * 8) * DIM + n0 + m] = val;
        csum += val;
        csq  += val * val;
    }
    // combine the two lane-halves (same column, complementary 8 rows)
    csum += __shfl_xor(csum, 16, 32);
    csq  += __shfl_xor(csq, 16, 32);
    if (hi == 0) {
        atomicAdd(&stats[n0 + m], csum);
        atomicAdd(&stats[DIM + n0 + m], csq);
    }
}

__global__ void finalize_stats_kernel(const float* __restrict__ stats,
                                      float* __restrict__ meaninv, float inv_n)
{
    int f = threadIdx.x;           // 64 threads
    float mean = stats[f] * inv_n;
    float var  = fmaxf(stats[DIM + f] * inv_n - mean * mean, 0.0f);
    meaninv[f]       = mean;
    meaninv[DIM + f] = rsqrtf(var + BN_EPS);
}

// ----------------------------------------- BatchNorm apply + graph pooling
// batch[] is sorted: accumulate a 16-node strip locally, one atomic per
// graph transition (~16x fewer contended atomics).
__global__ __launch_bounds__(256) void bn_pool_kernel(
    const float* __restrict__ z, const float* __restrict__ meaninv,
    const float* __restrict__ g, const float* __restrict__ b,
    const int* __restrict__ batch, float* __restrict__ h_out,
    float* __restrict__ pooled, int layer_off)
{
    const int STRIP = 16;
    long long gid = (long long)blockIdx.x * 256 + threadIdx.x;
    int f = (int)(gid & (DIM - 1));
    int node0 = (int)(gid >> 6) * STRIP;
    if (node0 >= N_NODES) return;
    float mean = meaninv[f], inv = meaninv[DIM + f];
    float gg = g[f], bb = b[f];
    int cur = batch[node0];
    float acc = 0.0f;
    int nend = min(node0 + STRIP, N_NODES);
    for (int node = node0; node < nend; ++node) {
        float val = (z[(size_t)node * DIM + f] - mean) * inv * gg + bb;
        h_out[(size_t)node * DIM + f] = val;
        int gr = batch[node];
        if (gr != cur) {
            atomicAdd(&pooled[(size_t)cur * EMB + layer_off + f], acc);
            acc = 0.0f; cur = gr;
        }
        acc += val;
    }
    atomicAdd(&pooled[(size_t)cur * EMB + layer_off + f], acc);
}

// ------------------------------------------- projection head (fused, WMMA)
// y = ReLU(pooled @ p_w1 + p_b1) @ p_w2 + p_b2    [1000 x 192]
__global__ __launch_bounds__(128) void proj_kernel(
    const float* __restrict__ pooled, const float* __restrict__ w1,
    const float* __restrict__ b1, const float* __restrict__ w2,
    const float* __restrict__ b2, float* __restrict__ out, int n_rows)
{
    const int EP = EMB + 4;
    __shared__ float lx[16 * (EMB + 4)];
    __shared__ float lz[16 * (EMB + 4)];
    const int tid = threadIdx.x, lane = tid & 31, wave = tid >> 5;
    const int row0 = blockIdx.x * 16;

    for (int i = tid; i < 16 * EMB; i += 128) {
        int r = i / EMB, c = i % EMB;
        int rr = row0 + r;
        lx[r * EP + c] = (rr < n_rows) ? pooled[(size_t)rr * EMB + c] : 0.0f;
    }
    __syncthreads();

    const int m = lane & 15, hi = lane >> 4;
    const int k0 = hi ? 2 : 0, k1 = hi ? 3 : 1;

    for (int t = 0; t < 3; ++t) {              // 12 N-tiles / 4 waves
        int n0 = (wave * 3 + t) * 16;
        v8f c1 = {};
        for (int k = 0; k < EMB; k += 4) {
            v2f a, b;
            a.x = lx[m * EP + k + k0];
            a.y = lx[m * EP + k + k1];
            b.x = w1[(k + k0) * EMB + n0 + m];
            b.y = w1[(k + k1) * EMB + n0 + m];
            c1 = __builtin_amdgcn_wmma_f32_16x16x4_f32(false, a, false, b,
                                                       (short)0, c1, false, false);
        }
        float bias = b1[n0 + m];
#pragma unroll
        for (int v = 0; v < 8; ++v)
            lz[(v + hi * 8) * EP + n0 + m] = fmaxf(c1[v] + bias, 0.0f);
    }
    __syncthreads();

    for (int t = 0; t < 3; ++t) {
        int n0 = (wave * 3 + t) * 16;
        v8f c2 = {};
        for (int k = 0; k < EMB; k += 4) {
            v2f a, b;
            a.x = lz[m * EP + k + k0];
            a.y = lz[m * EP + k + k1];
            b.x = w2[(k + k0) * EMB + n0 + m];
            b.y = w2[(k + k1) * EMB + n0 + m];
            c2 = __builtin_amdgcn_wmma_f32_16x16x4_f32(false, a, false, b,
                                                       (short)0, c2, false, false);
        }
        float bias = b2[n0 + m];
#pragma unroll
        for (int v = 0; v < 8; ++v) {
            int rr = row0 + v + hi * 8;
            if (rr < n_rows) out[(size_t)rr * EMB + n0 + m] = c2[v] + bias;
        }
    }
}

// ------------------------------------------------------------------ driver
extern "C" void kernel_launch(void* const* d_in, const int* in_sizes, int n_in,
                              void* d_out, int out_size, void* d_ws, size_t ws_size,
                              hipStream_t stream)
{
    const float* x      = (const float*)d_in[0];
    const int*   src    = (const int*)d_in[1];               // edge_index[0]
    const int*   dst    = ((const int*)d_in[1]) + N_EDGES;   // edge_index[1]
    const int*   batch  = (const int*)d_in[2];
    // d_in[3] = num_graphs (==1000, fixed by problem constants)
    const float* cw1[3] = {(const float*)d_in[4],  (const float*)d_in[10], (const float*)d_in[16]};
    const float* cb1[3] = {(const float*)d_in[5],  (const float*)d_in[11], (const float*)d_in[17]};
    const float* cw2[3] = {(const float*)d_in[6],  (const float*)d_in[12], (const float*)d_in[18]};
    const float* cb2[3] = {(const float*)d_in[7],  (const float*)d_in[13], (const float*)d_in[19]};
    const float* bng[3] = {(const float*)d_in[8],  (const float*)d_in[14], (const float*)d_in[20]};
    const float* bnb[3] = {(const float*)d_in[9],  (const float*)d_in[15], (const float*)d_in[21]};
    const float* pw1 = (const float*)d_in[22];
    const float* pb1 = (const float*)d_in[23];
    const float* pw2 = (const float*)d_in[24];
    const float* pb2 = (const float*)d_in[25];
    float* y = (float*)d_out;

    // workspace layout (all 256B aligned)
    char* ws = (char*)d_ws;
    float* agg     = (float*)(ws);                                     // 100000*128 f32
    float* zbuf    = (float*)(ws + (size_t)51200000);                  // 100000*64
    float* hbuf    = (float*)(ws + (size_t)51200000 + 25600000);       // 100000*64
    float* stats   = (float*)(ws + (size_t)51200000 + 2*25600000);     // 128 f32
    float* meaninv = stats + 128;                                      // 128 f32
    float* pooled  = meaninv + 128;                                    // 1000*192

    const float inv_n = 1.0f / (float)N_NODES;

    zero_kernel<<<(GRAPHS * EMB + 255) / 256, 256, 0, stream>>>(pooled, GRAPHS * EMB);

    for (int l = 0; l < 3; ++l) {
        const int FIN = (l == 0) ? FEAT : DIM;
        const float* hin = (l == 0) ? x : hbuf;
        const long long elems = (long long)N_NODES * FIN;

        // agg = h (self term), then agg[dst] += h[src]
        copy4_kernel<<<(int)((elems / 4 + 255) / 256), 256, 0, stream>>>(
            (const float4*)hin, (float4*)agg, (int)(elems / 4));
        const long long etotal = (long long)N_EDGES * FIN;
        if (FIN == FEAT)
            scatter_add_kernel<FEAT><<<(int)((etotal + 255) / 256), 256, 0, stream>>>(
                hin, src, dst, agg, etotal);
        else
            scatter_add_kernel<DIM><<<(int)((etotal + 255) / 256), 256, 0, stream>>>(
                hin, src, dst, agg, etotal);

        zero_kernel<<<1, 128, 0, stream>>>(stats, 128);
        if (FIN == FEAT)
            gin_mlp_kernel<FEAT><<<N_NODES / 16, 128, 0, stream>>>(
                agg, cw1[l], cb1[l], cw2[l], cb2[l], zbuf, stats);
        else
            gin_mlp_kernel<DIM><<<N_NODES / 16, 128, 0, stream>>>(
                agg, cw1[l], cb1[l], cw2[l], cb2[l], zbuf, stats);

        finalize_stats_kernel<<<1, DIM, 0, stream>>>(stats, meaninv, inv_n);

        const int strips = N_NODES / 16;   // 6250
        bn_pool_kernel<<<(strips * DIM + 255) / 256, 256, 0, stream>>>(
            zbuf, meaninv, bng[l], bnb[l], batch, hbuf, pooled, l * DIM);
    }

    proj_kernel<<<(GRAPHS + 15) / 16, 128, 0, stream>>>(
        pooled, pw1, pb1, pw2, pb2, y, GRAPHS);
}